// LTSB_CLS_49907519979772
// MI455X (gfx1250) — compile-verified
//
#include <hip/hip_runtime.h>
#include <hip/hip_bf16.h>
#include <math.h>

// ---------------- problem constants ----------------
#define B_SZ     1024
#define DIMV     128
#define DIM_MLP  2048
#define NCLS     1000
#define BUFN     16
#define MOM      0.999f
#define BN_EPS   1e-5f

#define KCH  32     // K-chunk per LDS stage (8 wmma k-steps)
#define LDSP 36     // padded LDS row stride in floats: 144B = 16B-aligned, 16-bank spread

typedef __attribute__((ext_vector_type(2))) float v2f;
typedef __attribute__((ext_vector_type(8))) float v8f;

// ---- CDNA5 async global->LDS DMA (ASYNCcnt-tracked), per-lane 16B copy ----
__device__ __forceinline__ void async_ld_b128(const float* gaddr, unsigned lds_off)
{
    asm volatile("global_load_async_to_lds_b128 %0, %1, off"
                 :: "v"(lds_off), "v"(gaddr) : "memory");
}
__device__ __forceinline__ void wait_async0()
{
    asm volatile("s_wait_asynccnt 0x0" ::: "memory");
}
__device__ __forceinline__ unsigned lds_off32(const void* p)
{
    // flat LDS addresses carry the LDS byte offset in addr[31:0]
    return (unsigned)(size_t)p;
}

// ============================================================
// NT GEMM with fused dual-source B (momentum combine) + bias:
//   C[M,N] = A[M,K] * (cb0*B0 + cb1*B1)[N,K]^T + (cb0*bias0 + cb1*bias1)
// Block: 256 threads = 8 waves; macro tile 128(M) x 64(N); K chunk 32.
// Each wave register-tiles 2x2 WMMA tiles (32x32) -> 32 WMMA per chunk.
// A tile: async DMA to double-buffered LDS; B tile: reg-stage + combine (DS store).
// ============================================================
__global__ __launch_bounds__(256) void gemm_nt_wmma(
    const float* __restrict__ A, const float* __restrict__ B0,
    const float* __restrict__ B1, float cb0, float cb1,
    const float* __restrict__ bias0, const float* __restrict__ bias1,
    float* __restrict__ C, int Mdim, int Ndim, int Kdim, int ldc)
{
    __shared__ __align__(16) float sA[2][128 * LDSP];
    __shared__ __align__(16) float sB[2][64 * LDSP];

    const int tid  = threadIdx.x;
    const int lane = tid & 31;
    const int wid  = tid >> 5;
    const int wm   = wid & 3;          // wave M position (0..3) -> rows wm*32
    const int wn   = wid >> 2;         // wave N position (0..1) -> cols wn*32
    const int m0   = blockIdx.y * 128;
    const int n0   = blockIdx.x * 64;

    const int m_l0 = (wm * 32 + (lane & 15)) * LDSP;
    const int m_l1 = m_l0 + 16 * LDSP;
    const int n_l0 = (wn * 32 + (lane & 15)) * LDSP;
    const int n_l1 = n_l0 + 16 * LDSP;
    const int kgrp = (lane >> 4) * 2;  // half-wave K-pair offset

    // ---- staging indices: row r = tid>>3 (0..31), 4-float col group ----
    const int r  = tid >> 3;           // 0..31
    const int c4 = (tid & 7) * 4;      // 0,4,...,28

    // A chunk 128x32 = 1024 x 16B pieces, 4/thread (rows r, r+32, r+64, r+96)
    const float* Arow[4];
    unsigned     aoff[2][4];
    #pragma unroll
    for (int i = 0; i < 4; ++i) {
        Arow[i] = A + (size_t)(m0 + r + 32 * i) * Kdim + c4;
        aoff[0][i] = lds_off32(&sA[0][(r + 32 * i) * LDSP + c4]);
        aoff[1][i] = lds_off32(&sA[1][(r + 32 * i) * LDSP + c4]);
    }
    // B chunk 64x32 = 512 x 16B pieces, 2/thread (rows r, r+32)
    const int  brow0 = n0 + r, brow1 = n0 + r + 32;
    const bool bv0 = brow0 < Ndim, bv1 = brow1 < Ndim;
    const float* B0r0 = B0 + (size_t)brow0 * Kdim + c4;
    const float* B1r0 = B1 + (size_t)brow0 * Kdim + c4;
    const float* B0r1 = B0 + (size_t)brow1 * Kdim + c4;
    const float* B1r1 = B1 + (size_t)brow1 * Kdim + c4;
    const int bo0 = r * LDSP + c4;            // LDS float index, row r
    const int bo1 = (r + 32) * LDSP + c4;     // LDS float index, row r+32

    v8f acc00 = {}, acc01 = {}, acc10 = {}, acc11 = {};

    const int nchunk = Kdim / KCH;

    // ---------------- prologue: chunk 0 into buffer 0 ----------------
    #pragma unroll
    for (int i = 0; i < 4; ++i) async_ld_b128(Arow[i], aoff[0][i]);
    {
        float4 p00 = {0,0,0,0}, p01 = {0,0,0,0}, p10 = {0,0,0,0}, p11 = {0,0,0,0};
        if (bv0) { p00 = *(const float4*)B0r0; p01 = *(const float4*)B1r0; }
        if (bv1) { p10 = *(const float4*)B0r1; p11 = *(const float4*)B1r1; }
        sB[0][bo0+0] = cb0*p00.x + cb1*p01.x; sB[0][bo0+1] = cb0*p00.y + cb1*p01.y;
        sB[0][bo0+2] = cb0*p00.z + cb1*p01.z; sB[0][bo0+3] = cb0*p00.w + cb1*p01.w;
        sB[0][bo1+0] = cb0*p10.x + cb1*p11.x; sB[0][bo1+1] = cb0*p10.y + cb1*p11.y;
        sB[0][bo1+2] = cb0*p10.z + cb1*p11.z; sB[0][bo1+3] = cb0*p10.w + cb1*p11.w;
    }
    wait_async0();
    __syncthreads();

    // ---------------- main loop: double-buffered ----------------
    for (int c = 0; c < nchunk; ++c) {
        const int cur = c & 1, nxt = cur ^ 1;
        const bool have_next = (c + 1) < nchunk;
        const int koff = (c + 1) * KCH;

        float4 p00 = {0,0,0,0}, p01 = {0,0,0,0}, p10 = {0,0,0,0}, p11 = {0,0,0,0};
        if (have_next) {
            // issue next A DMAs + next B global loads before computing
            #pragma unroll
            for (int i = 0; i < 4; ++i) async_ld_b128(Arow[i] + koff, aoff[nxt][i]);
            if (bv0) { p00 = *(const float4*)(B0r0 + koff); p01 = *(const float4*)(B1r0 + koff); }
            if (bv1) { p10 = *(const float4*)(B0r1 + koff); p11 = *(const float4*)(B1r1 + koff); }
        }

        // ---- compute: 8 k-steps x 4 WMMA (V_WMMA_F32_16X16X4_F32) ----
        const float* cA = &sA[cur][0];
        const float* cB = &sB[cur][0];
        #pragma unroll
        for (int ks = 0; ks < KCH / 4; ++ks) {
            const int kb = ks * 4 + kgrp;
            v2f a0 = { cA[m_l0 + kb], cA[m_l0 + kb + 1] };
            v2f a1 = { cA[m_l1 + kb], cA[m_l1 + kb + 1] };
            v2f b0 = { cB[n_l0 + kb], cB[n_l0 + kb + 1] };
            v2f b1 = { cB[n_l1 + kb], cB[n_l1 + kb + 1] };
            acc00 = __builtin_amdgcn_wmma_f32_16x16x4_f32(false, a0, false, b0, (short)0, acc00, false, false);
            acc01 = __builtin_amdgcn_wmma_f32_16x16x4_f32(false, a0, false, b1, (short)0, acc01, false, false);
            acc10 = __builtin_amdgcn_wmma_f32_16x16x4_f32(false, a1, false, b0, (short)0, acc10, false, false);
            acc11 = __builtin_amdgcn_wmma_f32_16x16x4_f32(false, a1, false, b1, (short)0, acc11, false, false);
        }

        if (have_next) {   // combine + DS-store next B tile (overlaps B load wait)
            sB[nxt][bo0+0] = cb0*p00.x + cb1*p01.x; sB[nxt][bo0+1] = cb0*p00.y + cb1*p01.y;
            sB[nxt][bo0+2] = cb0*p00.z + cb1*p01.z; sB[nxt][bo0+3] = cb0*p00.w + cb1*p01.w;
            sB[nxt][bo1+0] = cb0*p10.x + cb1*p11.x; sB[nxt][bo1+1] = cb0*p10.y + cb1*p11.y;
            sB[nxt][bo1+2] = cb0*p10.z + cb1*p11.z; sB[nxt][bo1+3] = cb0*p10.w + cb1*p11.w;
        }
        wait_async0();     // next A tile resident
        __syncthreads();   // all waves done reading cur, next buffers ready
    }

    // ---- epilogue: C layout VGPR v -> M = v + 8*(lane>=16) ----
    #pragma unroll
    for (int ti = 0; ti < 2; ++ti) {
        const int rbase = m0 + wm * 32 + ti * 16 + (lane >> 4) * 8;
        #pragma unroll
        for (int tj = 0; tj < 2; ++tj) {
            const int n = n0 + wn * 32 + tj * 16 + (lane & 15);
            if (n < Ndim) {
                float bv = 0.f;
                if (bias0) bv = cb0 * bias0[n] + cb1 * bias1[n];
                const v8f acc = (ti == 0) ? (tj == 0 ? acc00 : acc01)
                                          : (tj == 0 ? acc10 : acc11);
                #pragma unroll
                for (int v = 0; v < 8; ++v)
                    C[(size_t)(rbase + v) * ldc + n] = acc[v] + bv;
            }
        }
    }
}

// ============================================================
// BatchNorm (train-mode batch stats, biased var) + ReLU, in place.
// ============================================================
__global__ __launch_bounds__(256) void bn_relu_kernel(
    float* __restrict__ h,
    const float* __restrict__ g0, const float* __restrict__ g1,
    const float* __restrict__ be0, const float* __restrict__ be1,
    float c0, float c1)
{
    const int j = blockIdx.x;
    const int tid = threadIdx.x;
    float v[4], s = 0.f, s2 = 0.f;
    #pragma unroll
    for (int i = 0; i < 4; ++i) {
        v[i] = h[(size_t)(tid + i * 256) * DIM_MLP + j];
        s += v[i]; s2 += v[i] * v[i];
    }
    __shared__ float r1[256], r2[256];
    r1[tid] = s; r2[tid] = s2;
    __syncthreads();
    for (int st = 128; st > 0; st >>= 1) {
        if (tid < st) { r1[tid] += r1[tid + st]; r2[tid] += r2[tid + st]; }
        __syncthreads();
    }
    const float mu  = r1[0] * (1.f / 1024.f);
    const float var = r2[0] * (1.f / 1024.f) - mu * mu;
    const float gam = c0 * g0[j] + c1 * g1[j];
    const float bet = c0 * be0[j] + c1 * be1[j];
    const float sc  = gam * rsqrtf(var + BN_EPS);
    #pragma unroll
    for (int i = 0; i < 4; ++i) {
        const float o = sc * (v[i] - mu) + bet;
        h[(size_t)(tid + i * 256) * DIM_MLP + j] = fmaxf(o, 0.f);
    }
}

// ============================================================
// Row-wise L2 normalize [1024, 128] in place.
// ============================================================
__global__ __launch_bounds__(128) void l2norm_kernel(float* __restrict__ p)
{
    const int row = blockIdx.x, tid = threadIdx.x;
    const float x = p[(size_t)row * DIMV + tid];
    __shared__ float r[128];
    r[tid] = x * x;
    __syncthreads();
    for (int st = 64; st > 0; st >>= 1) {
        if (tid < st) r[tid] += r[tid + st];
        __syncthreads();
    }
    const float inv = 1.f / fmaxf(sqrtf(r[0]), 1e-12f);
    p[(size_t)row * DIMV + tid] = x * inv;
}

// ============================================================
// Per-row argmax (first-max tie-break, as jnp.argmax) + sigmoid prob.
// ============================================================
__global__ __launch_bounds__(256) void argmax_sig_kernel(
    const float* __restrict__ logits, const int* __restrict__ epoch,
    const int* __restrict__ target, int* __restrict__ pred,
    float* __restrict__ prob)
{
    const int row = blockIdx.x, tid = threadIdx.x;
    float best = -INFINITY; int bi = 0x7fffffff;
    for (int j = tid; j < NCLS; j += 256) {
        const float v = logits[(size_t)row * NCLS + j];
        if (v > best) { best = v; bi = j; }
    }
    __shared__ float sv[256]; __shared__ int si[256];
    sv[tid] = best; si[tid] = bi;
    __syncthreads();
    for (int st = 128; st > 0; st >>= 1) {
        if (tid < st) {
            const float ov = sv[tid + st]; const int oi = si[tid + st];
            if (ov > sv[tid] || (ov == sv[tid] && oi < si[tid])) {
                sv[tid] = ov; si[tid] = oi;
            }
        }
        __syncthreads();
    }
    if (tid == 0) {
        int p = si[0];
        if (epoch[0] < 0) p = target[row];
        const float l = logits[(size_t)row * NCLS + p];
        pred[row] = p;
        prob[row] = 1.f / (1.f + expf(-l));
    }
}

// ============================================================
// Gather from OLD buffers.
// ============================================================
__global__ __launch_bounds__(256) void gather_kernel(
    const int* __restrict__ pred, const int* __restrict__ rand_idx,
    const float* __restrict__ d_buf, const float* __restrict__ conf_buf,
    float* __restrict__ dout, float* __restrict__ conf_out)
{
    const int t = blockIdx.x * 256 + threadIdx.x;
    if (t >= B_SZ * DIMV) return;
    const int b = t >> 7, d = t & 127;
    const int p = pred[b], ri = rand_idx[b];
    dout[t] = d_buf[((size_t)p * BUFN + ri) * DIMV + d];
    if (d == 0) conf_out[b] = conf_buf[p * BUFN + ri];
}

// ============================================================
// Sequential circular scatter, one WAVE per class.
// ============================================================
__global__ __launch_bounds__(256) void scatter_kernel(
    const float* __restrict__ kvec, const float* __restrict__ prob,
    const int* __restrict__ pred, const float* __restrict__ d_buf,
    const float* __restrict__ conf_buf, const int* __restrict__ ptr,
    float* __restrict__ d_new, float* __restrict__ conf_new,
    float* __restrict__ ptr_new)
{
    const int lane = threadIdx.x & 31;
    const int c = blockIdx.x * 8 + (threadIdx.x >> 5);   // class id, wave-uniform
    if (c >= NCLS) return;

    const float4* src = (const float4*)(d_buf + (size_t)c * BUFN * DIMV);
    float4*       dst = (float4*)(d_new + (size_t)c * BUFN * DIMV);
    for (int t = lane; t < (BUFN * DIMV) / 4; t += 32) dst[t] = src[t];
    if (lane < BUFN) conf_new[c * BUFN + lane] = conf_buf[c * BUFN + lane];

    const int p = ptr[c];
    int cnt = 0;
    for (int b = 0; b < B_SZ; ++b) {
        if (pred[b] == c) {
            const int idx = (p + cnt) & (BUFN - 1);
            float4* drow = (float4*)(d_new + ((size_t)c * BUFN + idx) * DIMV);
            const float4* krow = (const float4*)(kvec + (size_t)b * DIMV);
            drow[lane] = krow[lane];
            if (lane == 0) conf_new[c * BUFN + idx] = prob[b];
            ++cnt;
        }
    }
    if (lane == 0) ptr_new[c] = (float)((p + cnt) & (BUFN - 1));
}

// ============================================================
extern "C" void kernel_launch(void* const* d_in, const int* in_sizes, int n_in,
                              void* d_out, int out_size, void* d_ws, size_t ws_size,
                              hipStream_t stream)
{
    const float* feat_q   = (const float*)d_in[0];
    const float* feat_k   = (const float*)d_in[1];
    const int*   target   = (const int*)  d_in[2];
    const int*   epoch    = (const int*)  d_in[3];
    const int*   rand_idx = (const int*)  d_in[4];
    const float* W1_q     = (const float*)d_in[5];
    const float* g_q      = (const float*)d_in[6];
    const float* be_q     = (const float*)d_in[7];
    const float* W2_q     = (const float*)d_in[8];
    const float* b2_q     = (const float*)d_in[9];
    const float* W1_k     = (const float*)d_in[10];
    const float* g_k      = (const float*)d_in[11];
    const float* be_k     = (const float*)d_in[12];
    const float* W2_k     = (const float*)d_in[13];
    const float* b2_k     = (const float*)d_in[14];
    const float* W_lin    = (const float*)d_in[15];
    const float* b_lin    = (const float*)d_in[16];
    const float* d_buf    = (const float*)d_in[17];
    const float* conf_buf = (const float*)d_in[18];
    const int*   ptr_in   = (const int*)  d_in[19];

    // output tuple laid out flat in return order
    float* out      = (float*)d_out;
    float* q_out    = out;                                  // [1024,128]
    float* k_out    = q_out    + B_SZ * DIMV;               // [1024,128]
    float* dout     = k_out    + B_SZ * DIMV;               // [1024,128]
    float* logits   = dout     + B_SZ * DIMV;               // [1024,1000]
    float* conf_out = logits   + B_SZ * NCLS;               // [1024]
    float* d_new    = conf_out + B_SZ;                      // [1000,16,128]
    float* conf_new = d_new    + NCLS * BUFN * DIMV;        // [1000,16]
    float* ptr_new  = conf_new + NCLS * BUFN;               // [1000]

    // workspace: h_q, h_k [1024,2048] each, pred, prob
    float* ws   = (float*)d_ws;
    float* h_q  = ws;
    float* h_k  = ws + (size_t)B_SZ * DIM_MLP;
    int*   pred = (int*)(ws + 2 * (size_t)B_SZ * DIM_MLP);
    float* prob = ws + 2 * (size_t)B_SZ * DIM_MLP + B_SZ;

    const float m1 = MOM, m0 = 1.f - MOM;
    const dim3 blk(256);

    // 1) h = feat @ W1^T  (q path; k path with momentum-fused weights)
    const dim3 g1((DIM_MLP + 63) / 64, B_SZ / 128);
    gemm_nt_wmma<<<g1, blk, 0, stream>>>(feat_q, W1_q, W1_q, 1.f, 0.f,
                                         nullptr, nullptr, h_q,
                                         B_SZ, DIM_MLP, DIM_MLP, DIM_MLP);
    gemm_nt_wmma<<<g1, blk, 0, stream>>>(feat_k, W1_k, W1_q, m1, m0,
                                         nullptr, nullptr, h_k,
                                         B_SZ, DIM_MLP, DIM_MLP, DIM_MLP);

    // 2) BatchNorm(train) + ReLU in place
    bn_relu_kernel<<<DIM_MLP, blk, 0, stream>>>(h_q, g_q, g_q, be_q, be_q, 1.f, 0.f);
    bn_relu_kernel<<<DIM_MLP, blk, 0, stream>>>(h_k, g_k, g_q, be_k, be_q, m1, m0);

    // 3) z = h @ W2^T + b2  (into output q/k regions)
    const dim3 g2((DIMV + 63) / 64, B_SZ / 128);
    gemm_nt_wmma<<<g2, blk, 0, stream>>>(h_q, W2_q, W2_q, 1.f, 0.f,
                                         b2_q, b2_q, q_out,
                                         B_SZ, DIMV, DIM_MLP, DIMV);
    gemm_nt_wmma<<<g2, blk, 0, stream>>>(h_k, W2_k, W2_q, m1, m0,
                                         b2_k, b2_q, k_out,
                                         B_SZ, DIMV, DIM_MLP, DIMV);

    // 4) logits = feat_q @ W_lin^T + b_lin
    const dim3 g3((NCLS + 63) / 64, B_SZ / 128);
    gemm_nt_wmma<<<g3, blk, 0, stream>>>(feat_q, W_lin, W_lin, 1.f, 0.f,
                                         b_lin, b_lin, logits,
                                         B_SZ, NCLS, DIM_MLP, NCLS);

    // 5) L2 normalize q and k
    l2norm_kernel<<<B_SZ, 128, 0, stream>>>(q_out);
    l2norm_kernel<<<B_SZ, 128, 0, stream>>>(k_out);

    // 6) pred = argmax(logits), prob = sigmoid(logits[b, pred])
    argmax_sig_kernel<<<B_SZ, blk, 0, stream>>>(logits, epoch, target, pred, prob);

    // 7) gather from old buffers
    gather_kernel<<<(B_SZ * DIMV + 255) / 256, blk, 0, stream>>>(
        pred, rand_idx, d_buf, conf_buf, dout, conf_out);

    // 8) sequential circular scatter (one wave per class)
    scatter_kernel<<<(NCLS + 7) / 8, blk, 0, stream>>>(
        k_out, prob, pred, d_buf, conf_buf, ptr_in, d_new, conf_new, ptr_new);
}